// inlwt_catone_65120294142328
// MI455X (gfx1250) — compile-verified
//
#include <hip/hip_runtime.h>

typedef __attribute__((ext_vector_type(2))) float v2f;
typedef __attribute__((ext_vector_type(8))) float v8f;

// Fixed lifting matrices (f32, same values as reference)
__constant__ float cM2[4][4] = {
  { 1.3968f, -0.2212f, -0.5412f,  1.3066f},
  { 0.2212f,  1.3968f, -1.3066f, -0.5412f},
  {-0.2212f, -1.3968f, -1.3066f, -0.5412f},
  {-1.3968f,  0.2212f, -0.5412f,  1.3066f}};
__constant__ float cM1[4][4] = {
  { 0.2166f, -0.0256f,  0.1213f,  0.0144f},
  { 0.0256f,  0.2166f,  0.0144f, -0.1213f},
  { 0.1213f, -0.0144f, -0.2166f, -0.0256f},
  {-0.0144f, -0.1213f,  0.0256f, -0.2166f}};

// Problem dims (fixed by the reference): in (16,256,128,128) f32 -> out (16,64,256,256) f32
constexpr int Bn = 16, Cc = 64, Hh = 128, Ww = 128;
constexpr long JSTRIDE = (long)Cc * Hh * Ww;   // stride between channel groups j -> j+1

__global__ __launch_bounds__(256) void inlwt_wmma_kernel(const float* __restrict__ in,
                                                         float* __restrict__ out) {
  const int lane = threadIdx.x & 31;
  const int wave = threadIdx.x >> 5;
  const int row  = lane & 15;   // A: output-component row / B,D: position column
  const int kp   = (lane >> 4) * 2;  // K pair handled by this lane: {0,1} or {2,3}

  // One wave per (b, cc, h) row
  const long rowId = (long)blockIdx.x * 8 + wave;
  const int h  = (int)(rowId % Hh);
  const int cc = (int)((rowId / Hh) % Cc);
  const int b  = (int)(rowId / ((long)Hh * Cc));

  // A_p (16x4, constant): A_p[m][k] = W[m][4p+k] = M1[m][p] * M2[p][4? no: ][col]
  // lane<16 holds K=kp..kp+1 per ISA 16x4 f32 A layout
  v2f a[4];
#pragma unroll
  for (int p = 0; p < 4; ++p) {
    const float m1 = (row < 4) ? cM1[row][p] : 0.0f;
    a[p][0] = m1 * cM2[p][kp];
    a[p][1] = m1 * cM2[p][kp + 1];
  }

  const int hm = (h + Hh - 1) & (Hh - 1);  // roll(+1): neighbor is h-1 (wrapped)
  const float* p0 = in + ((long)b * 4 * Cc + cc) * ((long)Hh * Ww) + (long)kp * JSTRIDE;
  const float* p1 = p0 + JSTRIDE;

  float* orow0 = out + (((long)b * Cc + cc) * (2 * Hh) + 2 * h) * (2L * Ww);
  float* orow1 = orow0 + 2 * Ww;

  for (int it = 0; it < Ww / 16; ++it) {
    const int w   = it * 16 + row;
    const int wm  = (w + Ww - 1) & (Ww - 1);      // roll(+1): neighbor w-1 (wrapped)
    const int o00 = h  * Ww + w;                  // (h,   w)
    const int o10 = hm * Ww + w;                  // (h-1, w)
    const int o01 = h  * Ww + wm;                 // (h,   w-1)
    const int o11 = hm * Ww + wm;                 // (h-1, w-1)

    if (it + 1 < Ww / 16) {                       // stream next group's rows
      __builtin_prefetch(p0 + o00 + 16, 0, 3);
      __builtin_prefetch(p1 + o00 + 16, 0, 3);
    }

    v8f acc = {};
    v2f bv;
    bv[0] = p0[o00]; bv[1] = p1[o00];
    acc = __builtin_amdgcn_wmma_f32_16x16x4_f32(false, a[0], false, bv, (short)0, acc, false, false);
    bv[0] = p0[o10]; bv[1] = p1[o10];
    acc = __builtin_amdgcn_wmma_f32_16x16x4_f32(false, a[1], false, bv, (short)0, acc, false, false);
    bv[0] = p0[o01]; bv[1] = p1[o01];
    acc = __builtin_amdgcn_wmma_f32_16x16x4_f32(false, a[2], false, bv, (short)0, acc, false, false);
    bv[0] = p0[o11]; bv[1] = p1[o11];
    acc = __builtin_amdgcn_wmma_f32_16x16x4_f32(false, a[3], false, bv, (short)0, acc, false, false);

    // D rows 0..3 live in acc[0..3] on lanes 0..15 (N = lane = position)
    // comp m = 2p+q -> out(2h+p, 2w+q): pack pairs into b64 stores
    if (lane < 16) {
      v2f r0; r0[0] = acc[0]; r0[1] = acc[1];
      v2f r1; r1[0] = acc[2]; r1[1] = acc[3];
      *(v2f*)(orow0 + 2 * w) = r0;
      *(v2f*)(orow1 + 2 * w) = r1;
    }
  }
}

extern "C" void kernel_launch(void* const* d_in, const int* in_sizes, int n_in,
                              void* d_out, int out_size, void* d_ws, size_t ws_size,
                              hipStream_t stream) {
  (void)in_sizes; (void)n_in; (void)out_size; (void)d_ws; (void)ws_size;
  const float* in = (const float*)d_in[0];
  float* out = (float*)d_out;
  const int blocks = Bn * Cc * Hh / 8;  // 8 waves per 256-thread block, 1 row each
  inlwt_wmma_kernel<<<blocks, 256, 0, stream>>>(in, out);
}